// KNN_dist_13898514170054
// MI455X (gfx1250) — compile-verified
//
#include <hip/hip_runtime.h>
#include <hip/hip_bf16.h>

// Problem constants (match reference)
#define B_ 8
#define N_ 4096
#define C_ 128
#define K_ 16

typedef __attribute__((ext_vector_type(2))) float v2f;
typedef __attribute__((ext_vector_type(8))) float v8f;

__device__ __forceinline__ float coord3(float x, float y, float z, int k) {
    // k in {0,1,2,3}: x, y, z, 0-pad (K dimension padded to 4 for WMMA)
    return (k == 0) ? x : ((k == 1) ? y : ((k == 2) ? z : 0.0f));
}

// Stable sorted insert into ascending (dist, idx) top-16 kept in registers.
// Strict '<' placement => equal distances keep earlier-inserted (lower index)
// entries first, matching jax.lax.top_k tie ordering.
__device__ __forceinline__ void insert16(float d, int ci,
                                         float (&bd)[16], int (&bi)[16]) {
    if (d < bd[15]) {
        bool placed = false;
        #pragma unroll
        for (int t = 15; t >= 1; --t) {
            const bool sh  = (!placed) && (d < bd[t - 1]);
            const bool put = (!placed) && (!sh);
            if (sh)  { bd[t] = bd[t - 1]; bi[t] = bi[t - 1]; }
            if (put) { bd[t] = d; bi[t] = ci; placed = true; }
        }
        if (!placed) { bd[0] = d; bi[0] = ci; }
    }
}

// -----------------------------------------------------------------------------
// Kernel 1: one wave32 per (batch, 16-query tile, candidate segment).
// WMMA f32 16x16x4 Gram tiles -> distances -> LDS transpose -> per-lane
// register-resident stable top-16 over this segment's candidate range.
// Writes sorted partial (dist, idx) lists to workspace.
// -----------------------------------------------------------------------------
__global__ void __launch_bounds__(32)
knn_topk_partial(const float* __restrict__ verts, float* __restrict__ part_d,
                 int* __restrict__ part_i, int S) {
    const int wave = blockIdx.x;                  // 0 .. B*(N/16)*S-1
    const int seg  = wave % S;
    const int tile = (wave / S) % (N_ / 16);
    const int b    = wave / (S * (N_ / 16));
    const int q0   = tile * 16;
    const int lane  = threadIdx.x;                // wave32
    const int col   = lane & 15;
    const int khalf = (lane < 16) ? 0 : 2;        // A/B frag K offset per half
    const int mbase = (lane < 16) ? 0 : 8;        // C/D row offset per half

    const float* V = verts + (size_t)b * N_ * 3;

    __shared__ float s_sqq[16];
    __shared__ float s_tile[16][17];              // +1 pad: conflict-free

    // A fragment: 16 query vertices x K=4 (x,y,z,0).
    // ISA 7.12.2 (32-bit A 16x4): lane<16 holds K=0,1; lane>=16 holds K=2,3.
    const int qm = q0 + col;
    const float qx = V[qm * 3 + 0], qy = V[qm * 3 + 1], qz = V[qm * 3 + 2];
    v2f afrag;
    afrag.x = coord3(qx, qy, qz, khalf + 0);
    afrag.y = coord3(qx, qy, qz, khalf + 1);
    if (lane < 16) s_sqq[lane] = qx * qx + qy * qy + qz * qz;
    __syncthreads();

    float sq_m[8];
    #pragma unroll
    for (int r = 0; r < 8; ++r) sq_m[r] = s_sqq[mbase + r];

    float bd[16];
    int   bi[16];
    #pragma unroll
    for (int t = 0; t < 16; ++t) { bd[t] = 3.0e38f; bi[t] = 0; }

    const int segLen = N_ / S;
    const int cbase  = seg * segLen;
    const int ntiles = segLen / 16;

    for (int ct = 0; ct < ntiles; ++ct) {
        const int c0 = cbase + ct * 16;
        const int cm = c0 + col;
        const float cx = V[cm * 3 + 0], cy = V[cm * 3 + 1], cz = V[cm * 3 + 2];
        // B fragment (4x16), mirror of A layout.
        v2f bfrag;
        bfrag.x = coord3(cx, cy, cz, khalf + 0);
        bfrag.y = coord3(cx, cy, cz, khalf + 1);
        const float sqc = cx * cx + cy * cy + cz * cz;

        v8f acc = {0.f, 0.f, 0.f, 0.f, 0.f, 0.f, 0.f, 0.f};
        // D = A x B: 16x16 query.candidate dot products on the matrix core.
        acc = __builtin_amdgcn_wmma_f32_16x16x4_f32(
            /*neg_a=*/false, afrag, /*neg_b=*/false, bfrag,
            /*c_mod=*/(short)0, acc, /*reuse_a=*/false, /*reuse_b=*/false);

        // dist[m][n] = |q_m|^2 + |c_n|^2 - 2 q_m.c_n ; transpose via LDS so
        // each owner lane (0..15) sees its full query row.
        #pragma unroll
        for (int r = 0; r < 8; ++r) {
            const float dist = fmaf(-2.0f, acc[r], sq_m[r] + sqc);
            s_tile[mbase + r][col] = dist;
        }
        __syncthreads();

        if (lane < 16) {
            #pragma unroll
            for (int j = 0; j < 16; ++j) {
                insert16(s_tile[lane][j], c0 + j, bd, bi);
            }
        }
        __syncthreads();
    }

    if (lane < 16) {
        const size_t q = (size_t)(b * N_ + q0 + lane);
        float* pd = part_d + (q * S + seg) * 16;
        int*   pi = part_i + (q * S + seg) * 16;
        #pragma unroll
        for (int t = 0; t < 16; ++t) { pd[t] = bd[t]; pi[t] = bi[t]; }
    }
}

// -----------------------------------------------------------------------------
// Kernel 2: one thread per point. Stable-merge the S sorted partial top-16
// lists (segment-major feed preserves global index-tie order), then gather
// neighbor vertices, run the 10->10->10->1 leaky-ReLU MLP (uniform-address
// weight reads -> scalar cache), and accumulate the weighted feature sum.
// -----------------------------------------------------------------------------
__global__ void __launch_bounds__(256)
mlp_gather_kernel(const float* __restrict__ feats, const float* __restrict__ verts,
                  const float* __restrict__ W1, const float* __restrict__ b1,
                  const float* __restrict__ W2, const float* __restrict__ b2,
                  const float* __restrict__ W3, const float* __restrict__ b3,
                  const float* __restrict__ part_d, const int* __restrict__ part_i,
                  int S, float* __restrict__ out) {
    const int g = blockIdx.x * blockDim.x + threadIdx.x;   // 0 .. B*N-1
    const int b = g / N_;
    const float* V = verts + (size_t)b * N_ * 3;
    const float* F = feats + (size_t)b * N_ * C_;

    // Merge S sorted partial lists -> final reference-ordered top-16.
    float bd[16];
    int   bi[16];
    #pragma unroll
    for (int t = 0; t < 16; ++t) { bd[t] = 3.0e38f; bi[t] = 0; }
    for (int seg = 0; seg < S; ++seg) {
        const float* pd = part_d + ((size_t)g * S + seg) * 16;
        const int*   pi = part_i + ((size_t)g * S + seg) * 16;
        #pragma unroll
        for (int t = 0; t < 16; ++t) insert16(pd[t], pi[t], bd, bi);
    }

    int nb[16];
    #pragma unroll
    for (int k = 0; k < 16; ++k) nb[k] = bi[k];

    const int i0 = nb[0];                                   // nearest = self
    const float v0x = V[i0 * 3 + 0], v0y = V[i0 * 3 + 1], v0z = V[i0 * 3 + 2];

    float w[16];
    #pragma unroll 1
    for (int k = 0; k < 16; ++k) {
        const int ik = nb[k];
        const float vx = V[ik * 3 + 0], vy = V[ik * 3 + 1], vz = V[ik * 3 + 2];
        const float dx = v0x - vx, dy = v0y - vy, dz = v0z - vz;
        const float n2 = dx * dx + dy * dy + dz * dz;
        const float nr = (n2 > 0.0f) ? sqrtf(n2) : 0.0f;    // _safe_norm
        const float x[10] = {v0x, v0y, v0z, vx, vy, vz, dx, dy, dz, nr};

        float h1[10];
        #pragma unroll
        for (int j = 0; j < 10; ++j) {
            float s = b1[j];
            #pragma unroll
            for (int i = 0; i < 10; ++i) s = fmaf(x[i], W1[i * 10 + j], s);
            h1[j] = (s >= 0.0f) ? s : 0.2f * s;
        }
        float h2[10];
        #pragma unroll
        for (int j = 0; j < 10; ++j) {
            float s = b2[j];
            #pragma unroll
            for (int i = 0; i < 10; ++i) s = fmaf(h1[i], W2[i * 10 + j], s);
            h2[j] = (s >= 0.0f) ? s : 0.2f * s;
        }
        float s = b3[0];
        #pragma unroll
        for (int i = 0; i < 10; ++i) s = fmaf(h2[i], W3[i], s);
        w[k] = s;
    }

    // out[c] = sum_k w[k] * F[nb[k]][c], two 64-channel chunks of float4 accs.
    float* o = out + (size_t)g * C_;
    #pragma unroll 1
    for (int ch = 0; ch < 2; ++ch) {
        float4 acc[16];
        #pragma unroll
        for (int t = 0; t < 16; ++t) acc[t] = make_float4(0.f, 0.f, 0.f, 0.f);
        #pragma unroll 1
        for (int k = 0; k < 16; ++k) {
            const float4* row = (const float4*)(F + (size_t)nb[k] * C_ + ch * 64);
            const float wk = w[k];
            #pragma unroll
            for (int t = 0; t < 16; ++t) {
                const float4 r = row[t];
                acc[t].x = fmaf(wk, r.x, acc[t].x);
                acc[t].y = fmaf(wk, r.y, acc[t].y);
                acc[t].z = fmaf(wk, r.z, acc[t].z);
                acc[t].w = fmaf(wk, r.w, acc[t].w);
            }
        }
        float4* op4 = (float4*)(o + ch * 64);
        #pragma unroll
        for (int t = 0; t < 16; ++t) op4[t] = acc[t];
    }
}

extern "C" void kernel_launch(void* const* d_in, const int* in_sizes, int n_in,
                              void* d_out, int out_size, void* d_ws, size_t ws_size,
                              hipStream_t stream) {
    const float* feats = (const float*)d_in[0];
    const float* verts = (const float*)d_in[1];
    const float* W1    = (const float*)d_in[2];
    const float* b1    = (const float*)d_in[3];
    const float* W2    = (const float*)d_in[4];
    const float* b2    = (const float*)d_in[5];
    const float* W3    = (const float*)d_in[6];
    const float* b3    = (const float*)d_in[7];
    float* out = (float*)d_out;

    // Workspace: part_d float[B*N*S*16] + part_i int[B*N*S*16] = 4 MiB * S.
    // S chosen deterministically from the fixed ws_size.
    const size_t per_seg = (size_t)B_ * N_ * 16 * (sizeof(float) + sizeof(int));
    int S = 1;
    if (ws_size >= 4 * per_seg) S = 4;
    else if (ws_size >= 2 * per_seg) S = 2;

    float* part_d = (float*)d_ws;
    int*   part_i = (int*)((char*)d_ws + (size_t)B_ * N_ * S * 16 * sizeof(float));

    // Kernel 1: 8 * 256 * S wave32s; each scans N/S candidates for 16 queries.
    knn_topk_partial<<<B_ * (N_ / 16) * S, 32, 0, stream>>>(verts, part_d, part_i, S);

    // Kernel 2: one thread per point (merge + MLP + weighted gather-sum).
    mlp_gather_kernel<<<(B_ * N_) / 256, 256, 0, stream>>>(
        feats, verts, W1, b1, W2, b2, W3, b3, part_d, part_i, S, out);
}